// VtraceAdvantageEstimation_3478923510390
// MI455X (gfx1250) — compile-verified
//
#include <hip/hip_runtime.h>
#include <hip/hip_bf16.h>
#include <math.h>

typedef __attribute__((ext_vector_type(2))) float v2f;
typedef __attribute__((ext_vector_type(8))) float v8f;

#define GAMMA_F 0.99f
#define KTRUNC 32            // truncation depth: term k ~ 0.01^k -> 1e-64 at k=32
#define KPAIRS (KTRUNC / 2)  // float2 ratio-power pairs
#define KSTEPS (KTRUNC / 4)  // number of 16x16x4 WMMA steps

// ---------------------------------------------------------------------------
// Phase 1: per-timestep scalars  ratio[t], delta[t], c[t]
// ---------------------------------------------------------------------------
__global__ void vt_prep(const float* __restrict__ rewards,
                        const float* __restrict__ nextv,
                        const float* __restrict__ dones,
                        const float* __restrict__ wlp,
                        const float* __restrict__ llp,
                        float* __restrict__ ratio,
                        float* __restrict__ delta,
                        float* __restrict__ cmul,
                        int T) {
    int t = blockIdx.x * blockDim.x + threadIdx.x;
    if (t >= T) return;
    float r  = fminf(1.0f, expf(llp[t] - wlp[t]));
    float nd = 1.0f - dones[t];
    delta[t] = r * (rewards[t] + nd * GAMMA_F * nextv[t]);
    cmul[t]  = nd * (1.0f - GAMMA_F);
    ratio[t] = r;
}

// ---------------------------------------------------------------------------
// Phase 2: A[t][k] = delta[t+k] * prod_{u=t}^{t+k-1} c[u]   (T x KTRUNC)
// ---------------------------------------------------------------------------
__global__ void vt_buildA(const float* __restrict__ delta,
                          const float* __restrict__ cmul,
                          float* __restrict__ A, int T) {
    int t = blockIdx.x * blockDim.x + threadIdx.x;
    if (t >= T) return;
    float p = 1.0f;
#pragma unroll
    for (int k = 0; k < KTRUNC; ++k) {
        int   s = t + k;
        float a = 0.0f;
        if (s < T) { a = delta[s] * p; p *= cmul[s]; }
        A[(size_t)t * KTRUNC + k] = a;
    }
}

// ---------------------------------------------------------------------------
// Phase 3: B pair table: Bpk[p][j] = (ratio[j]^(2p), ratio[j]^(2p+1))
// ---------------------------------------------------------------------------
__global__ void vt_buildB(const float* __restrict__ ratio,
                          float2* __restrict__ B, int T) {
    int j = blockIdx.x * blockDim.x + threadIdx.x;
    if (j >= T) return;
    float r = ratio[j];
    float cur = 1.0f;
#pragma unroll
    for (int p = 0; p < KPAIRS; ++p) {
        float2 v;
        v.x = cur;
        v.y = cur * r;
        B[(size_t)p * T + j] = v;
        cur = v.y * r;
    }
}

// ---------------------------------------------------------------------------
// Phase 4: out[t][j] = (A @ Bpow)[t][j] - values[j] via V_WMMA_F32_16X16X4_F32
// One wave per 16x16 output tile; 8 chained WMMAs cover K=32.
// ---------------------------------------------------------------------------
__global__ void __launch_bounds__(256)
vt_gemm(const float* __restrict__ A,   // [T][KTRUNC]
        const float* __restrict__ B,   // float2 pairs: [KPAIRS][T]
        const float* __restrict__ values,
        float* __restrict__ out, int T) {
    const int lane = threadIdx.x & 31;
    const int wave = threadIdx.x >> 5;
    const int wavesPerBlock = blockDim.x >> 5;
    const int tilesPerRow = T >> 4;

    int tile = blockIdx.x * wavesPerBlock + wave;
    int trow = tile / tilesPerRow;
    int tcol = tile - trow * tilesPerRow;
    int t0 = trow << 4;
    int j0 = tcol << 4;
    if (t0 >= T) return;   // whole-wave uniform; EXEC stays all-ones for WMMA

    const int h = lane >> 4;  // lane half: selects K=+0/+1 vs K=+2/+3 (A) and pair row (B)
    const int m = lane & 15;  // M (for A) / N (for B, C, D)

    v2f afrag[KSTEPS];
    v2f bfrag[KSTEPS];
    const v2f* Bp = (const v2f*)B;
#pragma unroll
    for (int s = 0; s < KSTEPS; ++s) {
        // A fragment, step s: lane holds A[t0+m][4s+2h], A[t0+m][4s+2h+1]
        afrag[s] = *(const v2f*)(A + (size_t)(t0 + m) * KTRUNC + 4 * s + 2 * h);
        // B fragment, step s: lane holds ratio[j0+m]^(4s+2h), ^(4s+2h+1)
        bfrag[s] = Bp[(size_t)(2 * s + h) * T + (j0 + m)];
    }

    v8f acc = {};
#pragma unroll
    for (int s = 0; s < KSTEPS; ++s) {
        acc = __builtin_amdgcn_wmma_f32_16x16x4_f32(
            /*neg_a=*/false, afrag[s], /*neg_b=*/false, bfrag[s],
            /*c_mod=*/(short)0, acc, /*reuse_a=*/false, /*reuse_b=*/false);
    }

    // Epilogue: D lane (h,m) VGPR r -> (M = r + 8h, N = m); subtract values[j].
    float vj = values[j0 + m];
#pragma unroll
    for (int r = 0; r < 8; ++r) {
        int row = t0 + r + 8 * h;
        __builtin_nontemporal_store(acc[r] - vj, out + (size_t)row * T + (j0 + m));
    }
}

// ---------------------------------------------------------------------------
extern "C" void kernel_launch(void* const* d_in, const int* in_sizes, int n_in,
                              void* d_out, int out_size, void* d_ws, size_t ws_size,
                              hipStream_t stream) {
    const float* rewards = (const float*)d_in[0];
    const float* values  = (const float*)d_in[1];
    const float* nextv   = (const float*)d_in[2];
    const float* dones   = (const float*)d_in[3];
    const float* wlp     = (const float*)d_in[4];
    const float* llp     = (const float*)d_in[5];
    float* out = (float*)d_out;
    const int T = in_sizes[0];  // 8192, multiple of 16

    // Workspace layout (floats): ratio[T] | delta[T] | c[T] | A[T*32] | Bpairs[16*T*2]
    float*  ws    = (float*)d_ws;
    float*  ratio = ws;
    float*  delta = ws + T;
    float*  cmul  = ws + 2 * (size_t)T;
    float*  A     = ws + 3 * (size_t)T;
    float2* B     = (float2*)(ws + 3 * (size_t)T + (size_t)T * KTRUNC);  // 8B aligned

    const int threads = 256;
    const int blocks  = (T + threads - 1) / threads;
    vt_prep  <<<blocks, threads, 0, stream>>>(rewards, nextv, dones, wlp, llp,
                                              ratio, delta, cmul, T);
    vt_buildA<<<blocks, threads, 0, stream>>>(delta, cmul, A, T);
    vt_buildB<<<blocks, threads, 0, stream>>>(ratio, B, T);

    const int tiles = (T / 16) * (T / 16);
    const int wavesPerBlock = 8;
    const int gblocks = (tiles + wavesPerBlock - 1) / wavesPerBlock;
    vt_gemm<<<gblocks, 32 * wavesPerBlock, 0, stream>>>(A, (const float*)B,
                                                        values, out, T);
}